// StableLipschitzNorm_68685116997814
// MI455X (gfx1250) — compile-verified
//
#include <hip/hip_runtime.h>
#include <hip/hip_bf16.h>

// StableLipschitzNorm for MI455X (gfx1250, wave32).
//
// Memory-bound (~880 MB moved, ~0.45 GFLOP): two streaming passes + a
// segment-max via u32 atomics on float bit patterns (valid: nj > 0).
//
// MI455X-specific data-movement choices:
//  * x_i / x_j / e_ij / out are single-touch streams  -> non-temporal
//    hints (TH_NT) so they don't evict L2.
//  * the 1.6 MB node-max table + 3.2 MB index are multi-touch -> default
//    RT policy, resident in the 192 MB L2 (pass-2 gather is L2-speed).
//  * 128-bit coalesced loads (4 x global_load_b128 per (e,h)).

#define SLN_EPS      1e-8f
#define SLN_ATT_NORM 2.0f
#define SLN_N_NODES  50000

typedef float f32x4 __attribute__((ext_vector_type(4)));

// ---------------------------------------------------------------- pass 0
__global__ void sln_init_max(unsigned int* __restrict__ maxbuf, int n) {
    int t = blockIdx.x * blockDim.x + threadIdx.x;
    if (t < n) maxbuf[t] = 0u;  // bit pattern of +0.0f
}

// --------------------------------------- helper: ||x[t,:16]||^2, NT loads
__device__ __forceinline__ float sln_sumsq16_nt(const float* __restrict__ x,
                                                size_t t) {
    const f32x4* p = reinterpret_cast<const f32x4*>(x) + t * 4;
    f32x4 a = __builtin_nontemporal_load(p + 0);
    f32x4 b = __builtin_nontemporal_load(p + 1);
    f32x4 c = __builtin_nontemporal_load(p + 2);
    f32x4 d = __builtin_nontemporal_load(p + 3);
    float s = a.x * a.x + a.y * a.y + a.z * a.z + a.w * a.w;
    s += b.x * b.x + b.y * b.y + b.z * b.z + b.w * b.w;
    s += c.x * c.x + c.y * c.y + c.z * c.z + c.w * c.w;
    s += d.x * d.x + d.y * d.y + d.z * d.z + d.w * d.w;
    return s;
}

// ---------------------------------------------------------------- pass 1
// Per (e,h): nj = sqrt(sum x_j^2) + eps; atomic max into maxbuf[node*H+h].
// nj > 0 so unsigned-int compare == float compare -> global_atomic_max_u32.
__global__ void sln_scatter_max(const float* __restrict__ x_j,
                                const int* __restrict__ index,
                                unsigned int* __restrict__ maxbuf,
                                int EH, int hshift) {
    int t = blockIdx.x * blockDim.x + threadIdx.x;
    if (t >= EH) return;
    float nj = __builtin_sqrtf(sln_sumsq16_nt(x_j, (size_t)t)) + SLN_EPS;
    int e = t >> hshift;                 // H is a power of two (8)
    int h = t & ((1 << hshift) - 1);
    int node = index[e];                 // RT: index stays L2-resident
    atomicMax(maxbuf + ((size_t)node << hshift) + h, __float_as_uint(nj));
}

// ---------------------------------------------------------------- pass 2
// Per (e,h): ni = sqrt(sum x_i^2) + eps; gather node max (L2-resident);
// out = clip(e_ij / (2*(ni + (max+eps)) + eps), -10, 10).
__global__ void sln_finalize(const float* __restrict__ e_ij,
                             const float* __restrict__ x_i,
                             const int* __restrict__ index,
                             const unsigned int* __restrict__ maxbuf,
                             float* __restrict__ out,
                             int EH, int hshift) {
    int t = blockIdx.x * blockDim.x + threadIdx.x;
    if (t >= EH) return;
    float ni = __builtin_sqrtf(sln_sumsq16_nt(x_i, (size_t)t)) + SLN_EPS;
    int e = t >> hshift;
    int h = t & ((1 << hshift) - 1);
    float mx = __uint_as_float(maxbuf[((size_t)index[e] << hshift) + h]) + SLN_EPS;
    float denom = SLN_ATT_NORM * (ni + mx) + SLN_EPS;
    float r = __builtin_nontemporal_load(e_ij + t) / denom;
    r = fminf(10.0f, fmaxf(-10.0f, r));
    __builtin_nontemporal_store(r, out + t);
}

extern "C" void kernel_launch(void* const* d_in, const int* in_sizes, int n_in,
                              void* d_out, int out_size, void* d_ws, size_t ws_size,
                              hipStream_t stream) {
    const float* e_ij = (const float*)d_in[0];  // [E, H]
    const float* x_i  = (const float*)d_in[1];  // [E, H, 16]
    const float* x_j  = (const float*)d_in[2];  // [E, H, 16]
    const int*   idx  = (const int*)d_in[3];    // [E]

    const int E  = in_sizes[3];
    const int H  = in_sizes[0] / E;   // 8 (power of two per reference)
    const int EH = in_sizes[0];       // E*H
    int hshift = 0;
    while ((1 << hshift) < H) ++hshift;
    float* out = (float*)d_out;

    unsigned int* maxbuf = (unsigned int*)d_ws;      // N_NODES*H u32 = 1.6 MB
    const int nmax = SLN_N_NODES * H;

    const int BLK = 256;  // 8 wave32 waves per block
    sln_init_max<<<(nmax + BLK - 1) / BLK, BLK, 0, stream>>>(maxbuf, nmax);
    sln_scatter_max<<<(EH + BLK - 1) / BLK, BLK, 0, stream>>>(x_j, idx, maxbuf, EH, hshift);
    sln_finalize<<<(EH + BLK - 1) / BLK, BLK, 0, stream>>>(e_ij, x_i, idx, maxbuf, out, EH, hshift);
}